// GraphSAGE1_69286412419425
// MI455X (gfx1250) — compile-verified
//
#include <hip/hip_runtime.h>

#define D 128

typedef float v2f __attribute__((ext_vector_type(2)));
typedef float v8f __attribute__((ext_vector_type(8)));

// ---------------- degree ----------------
__global__ void sage_deg_kernel(const int* __restrict__ dst, float* deg, int E) {
    int e = blockIdx.x * blockDim.x + threadIdx.x;
    if (e < E) unsafeAtomicAdd(deg + dst[e], 1.0f);
}

__global__ void sage_invdeg_kernel(float* deg, int N) {
    int i = blockIdx.x * blockDim.x + threadIdx.x;
    if (i < N) {
        float d = deg[i];
        deg[i] = 1.0f / fmaxf(d, 1.0f);
    }
}

// ---------------- gather + scatter-add: one wave per edge ----------------
__global__ void sage_scatter_kernel(const float* __restrict__ x,
                                    const int* __restrict__ src,
                                    const int* __restrict__ dst,
                                    float* agg, int E) {
    int t = blockIdx.x * blockDim.x + threadIdx.x;
    int e = t >> 5;          // wave-uniform
    int lane = t & 31;
    if (e >= E) return;
    int s = src[e];
    int d = dst[e];
    float4 v = reinterpret_cast<const float4*>(x + (size_t)s * D)[lane];
    float* ap = agg + (size_t)d * D + lane * 4;
    unsafeAtomicAdd(ap + 0, v.x);
    unsafeAtomicAdd(ap + 1, v.y);
    unsafeAtomicAdd(ap + 2, v.z);
    unsafeAtomicAdd(ap + 3, v.w);
}

// ---------------- fused SAGE layer: out = act((agg*inv)@Wl^T + xin@Wr^T + bl)
// One wave computes a 16-row x 128-col output stripe with fp32 WMMA.
// out may alias agg (each wave reads its rows fully before writing them).
__global__ void sage_gemm_kernel(const float* agg,
                                 const float* __restrict__ xin,
                                 const float* __restrict__ invdeg,
                                 const float* __restrict__ Wl,
                                 const float* __restrict__ bl,
                                 const float* __restrict__ Wr,
                                 float* out, int N, int relu) {
    const int lane = threadIdx.x & 31;
    const int wave = threadIdx.x >> 5;
    const int rowBase = (blockIdx.x * (blockDim.x >> 5) + wave) * 16;
    if (rowBase >= N) return;                 // wave-uniform exit: EXEC stays all-1s

    // A-fragment lane mapping (16x4 f32): row = lane%16, k-pair = 2*(lane/16)
    const int mA   = lane & 15;
    const int kgrp = (lane >> 4) << 1;        // 0 or 2
    int row = rowBase + mA;
    if (row > N - 1) row = N - 1;             // clamp (no divergence)
    const float inv = invdeg[row];
    const float* aRow = agg + (size_t)row * D;
    const float* xRow = xin + (size_t)row * D;
    // B-fragment lane mapping (4x16 f32): col = lane%16, same k-pair
    const int nB = lane & 15;

    v8f acc[8] = {};

    for (int k0 = 0; k0 < D; k0 += 4) {
        v2f a;
        a[0] = aRow[k0 + kgrp]     * inv;
        a[1] = aRow[k0 + kgrp + 1] * inv;
#pragma unroll
        for (int t = 0; t < 8; ++t) {
            const float* wrow = Wl + (size_t)(t * 16 + nB) * D + k0 + kgrp;
            v2f b; b[0] = wrow[0]; b[1] = wrow[1];
            acc[t] = __builtin_amdgcn_wmma_f32_16x16x4_f32(
                false, a, false, b, (short)0, acc[t], false, false);
        }
        v2f a2;
        a2[0] = xRow[k0 + kgrp];
        a2[1] = xRow[k0 + kgrp + 1];
#pragma unroll
        for (int t = 0; t < 8; ++t) {
            const float* wrow = Wr + (size_t)(t * 16 + nB) * D + k0 + kgrp;
            v2f b; b[0] = wrow[0]; b[1] = wrow[1];
            acc[t] = __builtin_amdgcn_wmma_f32_16x16x4_f32(
                false, a2, false, b, (short)0, acc[t], false, false);
        }
    }

    // D layout: lane l, VGPR v holds C[v + 8*(l/16)][l%16]
    const int mBase = rowBase + ((lane >> 4) << 3);
#pragma unroll
    for (int t = 0; t < 8; ++t) {
        const int f = t * 16 + nB;
        const float bias = bl[f];
#pragma unroll
        for (int v = 0; v < 8; ++v) {
            const int m = mBase + v;
            if (m < N) {
                float val = acc[t][v] + bias;
                if (relu) val = fmaxf(val, 0.0f);
                out[(size_t)m * D + f] = val;
            }
        }
    }
}

// ---------------- output head: wave per node, 2 dot products of length 128
__global__ void sage_head_kernel(const float* __restrict__ h,
                                 const float* __restrict__ Wo,
                                 const float* __restrict__ bo,
                                 float* out, int N) {
    int t = blockIdx.x * blockDim.x + threadIdx.x;
    int node = t >> 5;       // wave-uniform
    int lane = t & 31;
    if (node >= N) return;
    float4 hv = reinterpret_cast<const float4*>(h + (size_t)node * D)[lane];
    float4 w0 = reinterpret_cast<const float4*>(Wo)[lane];
    float4 w1 = reinterpret_cast<const float4*>(Wo + D)[lane];
    float a0 = hv.x * w0.x + hv.y * w0.y + hv.z * w0.z + hv.w * w0.w;
    float a1 = hv.x * w1.x + hv.y * w1.y + hv.z * w1.z + hv.w * w1.w;
    for (int off = 16; off > 0; off >>= 1) {
        a0 += __shfl_xor(a0, off, 32);
        a1 += __shfl_xor(a1, off, 32);
    }
    if (lane == 0) {
        out[node * 2 + 0] = a0 + bo[0];
        out[node * 2 + 1] = a1 + bo[1];
    }
}

extern "C" void kernel_launch(void* const* d_in, const int* in_sizes, int n_in,
                              void* d_out, int out_size, void* d_ws, size_t ws_size,
                              hipStream_t stream) {
    const float* x   = (const float*)d_in[0];
    const int*   ei  = (const int*)d_in[1];
    const float* Wl1 = (const float*)d_in[2];
    const float* bl1 = (const float*)d_in[3];
    const float* Wr1 = (const float*)d_in[4];
    const float* Wl2 = (const float*)d_in[5];
    const float* bl2 = (const float*)d_in[6];
    const float* Wr2 = (const float*)d_in[7];
    const float* Wl3 = (const float*)d_in[8];
    const float* bl3 = (const float*)d_in[9];
    const float* Wr3 = (const float*)d_in[10];
    const float* Wo  = (const float*)d_in[11];
    const float* bo  = (const float*)d_in[12];
    float* out = (float*)d_out;

    const int N = in_sizes[0] / D;
    const int E = in_sizes[1] / 2;
    const int* src = ei;
    const int* dst = ei + E;

    // workspace layout: invdeg[N] | bufA[N*128] | bufB[N*128]
    float* ws = (float*)d_ws;
    size_t nAlign = ((size_t)N + 255) & ~(size_t)255;
    float* invdeg = ws;
    float* bufA = ws + nAlign;
    float* bufB = bufA + (size_t)N * D;

    const int TB = 256;
    const int degBlocks     = (E + TB - 1) / TB;
    const int invBlocks     = (N + TB - 1) / TB;
    const int scatterBlocks = (E * 32 + TB - 1) / TB;  // wave per edge
    const int gemmBlocks    = (N + 127) / 128;          // 8 waves * 16 rows per block
    const int headBlocks    = (N * 32 + TB - 1) / TB;   // wave per node

    // degrees (recomputed every call: deterministic, no cached state)
    hipMemsetAsync(invdeg, 0, (size_t)N * sizeof(float), stream);
    sage_deg_kernel<<<degBlocks, TB, 0, stream>>>(dst, invdeg, E);
    sage_invdeg_kernel<<<invBlocks, TB, 0, stream>>>(invdeg, N);

    // layer 1: x -> bufA
    hipMemsetAsync(bufA, 0, (size_t)N * D * sizeof(float), stream);
    sage_scatter_kernel<<<scatterBlocks, TB, 0, stream>>>(x, src, dst, bufA, E);
    sage_gemm_kernel<<<gemmBlocks, TB, 0, stream>>>(bufA, x, invdeg, Wl1, bl1, Wr1, bufA, N, 1);

    // layer 2: bufA -> bufB
    hipMemsetAsync(bufB, 0, (size_t)N * D * sizeof(float), stream);
    sage_scatter_kernel<<<scatterBlocks, TB, 0, stream>>>(bufA, src, dst, bufB, E);
    sage_gemm_kernel<<<gemmBlocks, TB, 0, stream>>>(bufB, bufA, invdeg, Wl2, bl2, Wr2, bufB, N, 1);

    // layer 3: bufB -> bufA (no relu)
    hipMemsetAsync(bufA, 0, (size_t)N * D * sizeof(float), stream);
    sage_scatter_kernel<<<scatterBlocks, TB, 0, stream>>>(bufB, src, dst, bufA, E);
    sage_gemm_kernel<<<gemmBlocks, TB, 0, stream>>>(bufA, bufB, invdeg, Wl3, bl3, Wr3, bufA, N, 0);

    // head: bufA -> out [N,2]
    sage_head_kernel<<<headBlocks, TB, 0, stream>>>(bufA, Wo, bo, out, N);

    (void)n_in; (void)out_size; (void)ws_size;
}